// GAT_34256659153436
// MI455X (gfx1250) — compile-verified
//
#include <hip/hip_runtime.h>
#include <hip/hip_bf16.h>
#include <math.h>

typedef __attribute__((ext_vector_type(2))) float v2f;
typedef __attribute__((ext_vector_type(8))) float v8f;

#define N_NODES   50000
#define N_EDGES   800000
#define F_IN      128
#define HC        128      // HEADS * HID
#define HEADS     4
#define HID       32
#define OUT_C     32
#define NEG_SLOPE 0.2f

// ---------------------------------------------------------------------------
// WMMA fp32 GEMM: C[N,P] = A[N,K] @ B[K,P] (+ bias[P] if bias != nullptr)
// One wave32 computes one 16x16 tile.  K % 4 == 0, N % 16 == 0, P % 16 == 0.
//
// A-fragment (16x4 f32, ISA 7.12.2): lanes 0-15 carry M=lane, K={0,1} in
// v[0],v[1]; lanes 16-31 carry M=lane-16, K={2,3}.  B-fragment (4x16) is the
// mirrored layout (N striped across lanes, K pairs split across lane halves).
// C/D: VGPR i -> M = i (lanes 0-15) / i+8 (lanes 16-31), N = lane % 16.
// ---------------------------------------------------------------------------
__global__ void gemm16x16_wmma_f32(const float* __restrict__ A,
                                   const float* __restrict__ B,
                                   const float* __restrict__ bias,
                                   float* __restrict__ C,
                                   int N, int K, int P) {
    const int wave = threadIdx.x >> 5;
    const int lane = threadIdx.x & 31;
    const int tile = blockIdx.x * (blockDim.x >> 5) + wave;
    const int ptiles = P >> 4;
    const int ntiles = N >> 4;
    if (tile >= ntiles * ptiles) return;
    const int row0 = (tile / ptiles) << 4;
    const int col0 = (tile % ptiles) << 4;
    const int half = lane >> 4;   // 0 -> K base 0, 1 -> K base 2
    const int r    = lane & 15;

    v8f acc = {};
    const float* arow = A + (size_t)(row0 + r) * K;
    for (int k0 = 0; k0 < K; k0 += 4) {
        const int ka = k0 + 2 * half;
        v2f a = *(const v2f*)(arow + ka);            // K, K+1 for this half
        v2f b;
        b.x = B[(size_t)(ka + 0) * P + col0 + r];
        b.y = B[(size_t)(ka + 1) * P + col0 + r];
        acc = __builtin_amdgcn_wmma_f32_16x16x4_f32(
            /*neg_a=*/false, a, /*neg_b=*/false, b,
            /*c_mod=*/(short)0, acc, /*reuse_a=*/false, /*reuse_b=*/false);
    }

    const float badd = bias ? bias[col0 + r] : 0.0f;
#pragma unroll
    for (int i = 0; i < 8; ++i) {
        const int row = row0 + i + half * 8;
        C[(size_t)row * P + col0 + r] = acc[i] + badd;
    }
}

// ---------------------------------------------------------------------------
// Attention logits: als[n,h] = dot(hfeat[n,h,:], aS[h,:]);  ald likewise.
// One thread per (node, head).
// ---------------------------------------------------------------------------
__global__ void attn_logits(const float* __restrict__ h,
                            const float* __restrict__ aS,
                            const float* __restrict__ aD,
                            float* __restrict__ als,
                            float* __restrict__ ald, int N) {
    const int i = blockIdx.x * blockDim.x + threadIdx.x;
    if (i >= N * HEADS) return;
    const int n  = i >> 2;
    const int hd = i & 3;
    const float* hp = h + (size_t)n * HC + hd * HID;
    const float* as = aS + hd * HID;
    const float* ad = aD + hd * HID;
    float ss = 0.f, sd = 0.f;
#pragma unroll 8
    for (int c = 0; c < HID; ++c) {
        const float v = hp[c];
        ss += v * as[c];
        sd += v * ad[c];
    }
    als[i] = ss;
    ald[i] = sd;
}

// ---------------------------------------------------------------------------
// Zero attention accumulator, softmax max (ordered-uint) and denominator.
// ---------------------------------------------------------------------------
__global__ void init_buffers(float* __restrict__ att, unsigned* __restrict__ m,
                             float* __restrict__ s, int natt, int nms) {
    const int i = blockIdx.x * blockDim.x + threadIdx.x;
    if (i < natt) att[i] = 0.0f;
    if (i < nms) { m[i] = 0u; s[i] = 0.0f; }   // 0u == encoding below -inf
}

// monotone float <-> orderable uint
__device__ __forceinline__ unsigned f2ord(float f) {
    unsigned u = __float_as_uint(f);
    return (u & 0x80000000u) ? ~u : (u | 0x80000000u);
}
__device__ __forceinline__ float ord2f(unsigned u) {
    return (u & 0x80000000u) ? __uint_as_float(u ^ 0x80000000u)
                             : __uint_as_float(~u);
}

// pass 1: e = leaky_relu(als[src]+ald[dst]); scatter max into m[dst]
__global__ void edge_max(const int* __restrict__ src, const int* __restrict__ dst,
                         const float* __restrict__ als, const float* __restrict__ ald,
                         float* __restrict__ ebuf, unsigned* __restrict__ m) {
    const int i = blockIdx.x * blockDim.x + threadIdx.x;
    if (i >= N_EDGES * HEADS) return;
    const int e  = i >> 2;
    const int hd = i & 3;
    const int sN = src[e], dN = dst[e];
    float v = als[sN * HEADS + hd] + ald[dN * HEADS + hd];
    v = (v > 0.f) ? v : NEG_SLOPE * v;
    ebuf[i] = v;
    atomicMax(&m[dN * HEADS + hd], f2ord(v));
}

// pass 2: ex = exp(e - m[dst]); scatter add into s[dst]
__global__ void edge_exp(const int* __restrict__ dst, const unsigned* __restrict__ m,
                         float* __restrict__ ebuf, float* __restrict__ s) {
    const int i = blockIdx.x * blockDim.x + threadIdx.x;
    if (i >= N_EDGES * HEADS) return;
    const int e  = i >> 2;
    const int hd = i & 3;
    const int dN = dst[e];
    const float ex = expf(ebuf[i] - ord2f(m[dN * HEADS + hd]));
    ebuf[i] = ex;
    atomicAdd(&s[dN * HEADS + hd], ex);
}

// pass 3: att[dst, c] += (ex / (s[dst]+eps)) * h[src, c], one thread per (edge, c)
__global__ void edge_aggr(const int* __restrict__ src, const int* __restrict__ dst,
                          const float* __restrict__ ebuf, const float* __restrict__ s,
                          const float* __restrict__ h, float* __restrict__ att) {
    const long long i = (long long)blockIdx.x * blockDim.x + threadIdx.x;
    if (i >= (long long)N_EDGES * HC) return;
    const int e  = (int)(i >> 7);
    const int c  = (int)(i & 127);
    const int hd = c >> 5;
    const int sN = src[e], dN = dst[e];
    const float alpha = ebuf[e * HEADS + hd] / (s[dN * HEADS + hd] + 1e-16f);
    atomicAdd(&att[(size_t)dN * HC + c], alpha * h[(size_t)sN * HC + c]);
}

// layers 0/1 epilogue: out = elu(att + b + skip)
__global__ void finish01(const float* __restrict__ att, const float* __restrict__ b,
                         const float* __restrict__ skip, float* __restrict__ out, int N) {
    const int i = blockIdx.x * blockDim.x + threadIdx.x;
    if (i >= N * HC) return;
    const int c = i & 127;
    const float v = att[i] + b[c] + skip[i];
    out[i] = (v > 0.f) ? v : (expf(v) - 1.0f);
}

// layer 2 epilogue: mean over heads + b2 + skip, then L2 normalize per node.
// One wave32 per node, lane == channel.
__global__ void finish2(const float* __restrict__ att, const float* __restrict__ b2,
                        const float* __restrict__ skip32, float* __restrict__ out, int N) {
    const int node = blockIdx.x * (blockDim.x >> 5) + (threadIdx.x >> 5);
    const int c    = threadIdx.x & 31;
    if (node >= N) return;
    const float* ap = att + (size_t)node * HC;
    float v = 0.25f * (ap[c] + ap[32 + c] + ap[64 + c] + ap[96 + c])
            + b2[c] + skip32[(size_t)node * OUT_C + c];
    float sq = v * v;
#pragma unroll
    for (int o = 16; o > 0; o >>= 1) sq += __shfl_xor(sq, o, 32);
    const float norm = sqrtf(sq);
    out[(size_t)node * OUT_C + c] = v / fmaxf(norm, 1e-12f);
}

// ---------------------------------------------------------------------------
extern "C" void kernel_launch(void* const* d_in, const int* in_sizes, int n_in,
                              void* d_out, int out_size, void* d_ws, size_t ws_size,
                              hipStream_t stream) {
    (void)in_sizes; (void)n_in; (void)out_size; (void)ws_size;

    const float* x   = (const float*)d_in[0];
    const int*   ei  = (const int*)d_in[1];
    const float* W0  = (const float*)d_in[2];
    const float* aS0 = (const float*)d_in[3];
    const float* aD0 = (const float*)d_in[4];
    const float* b0  = (const float*)d_in[5];
    const float* Ws0 = (const float*)d_in[6];
    const float* bs0 = (const float*)d_in[7];
    const float* W1  = (const float*)d_in[8];
    const float* aS1 = (const float*)d_in[9];
    const float* aD1 = (const float*)d_in[10];
    const float* b1  = (const float*)d_in[11];
    const float* Ws1 = (const float*)d_in[12];
    const float* bs1 = (const float*)d_in[13];
    const float* W2  = (const float*)d_in[14];
    const float* aS2 = (const float*)d_in[15];
    const float* aD2 = (const float*)d_in[16];
    const float* b2  = (const float*)d_in[17];
    const float* Ws2 = (const float*)d_in[18];
    const float* bs2 = (const float*)d_in[19];

    const int* src = ei;
    const int* dst = ei + N_EDGES;

    // workspace layout (floats)
    const size_t NF = (size_t)N_NODES * HC;
    float*    ws    = (float*)d_ws;
    float*    x_cur = ws;                 // N*128
    float*    h_lin = ws + NF;            // N*128
    float*    skipb = ws + 2 * NF;        // N*128 (layer2 uses first N*32)
    float*    att   = ws + 3 * NF;        // N*128
    float*    als   = ws + 4 * NF;        // N*4
    float*    ald   = als + (size_t)N_NODES * HEADS;
    unsigned* m     = (unsigned*)(ald + (size_t)N_NODES * HEADS);
    float*    s     = (float*)(m + (size_t)N_NODES * HEADS);
    float*    ebuf  = s + (size_t)N_NODES * HEADS;   // E*4

    auto gemm = [&](const float* A, const float* B, const float* bias,
                    float* C, int N, int K, int P) {
        const int tiles = (N / 16) * (P / 16);
        const int blocks = (tiles + 7) / 8;            // 8 waves / block
        gemm16x16_wmma_f32<<<blocks, 256, 0, stream>>>(A, B, bias, C, N, K, P);
    };

    auto run_edges = [&](const float* hfeat) {
        const int nInit = N_NODES * HC;
        init_buffers<<<(nInit + 255) / 256, 256, 0, stream>>>(
            att, m, s, nInit, N_NODES * HEADS);
        const int nEH = N_EDGES * HEADS;
        edge_max<<<(nEH + 255) / 256, 256, 0, stream>>>(src, dst, als, ald, ebuf, m);
        edge_exp<<<(nEH + 255) / 256, 256, 0, stream>>>(dst, m, ebuf, s);
        const long long tot = (long long)N_EDGES * HC;
        edge_aggr<<<(int)((tot + 255) / 256), 256, 0, stream>>>(
            src, dst, ebuf, s, hfeat, att);
    };

    const int nNH = N_NODES * HEADS;
    const int nNF = N_NODES * HC;

    // ---- layer 0 : F_IN=128 -> HC=128 (concat) -------------------------------
    gemm(x, W0, nullptr, h_lin, N_NODES, F_IN, HC);
    gemm(x, Ws0, bs0, skipb, N_NODES, F_IN, HC);
    attn_logits<<<(nNH + 255) / 256, 256, 0, stream>>>(h_lin, aS0, aD0, als, ald, N_NODES);
    run_edges(h_lin);
    finish01<<<(nNF + 255) / 256, 256, 0, stream>>>(att, b0, skipb, x_cur, N_NODES);

    // ---- layer 1 : HC -> HC (concat) ----------------------------------------
    gemm(x_cur, W1, nullptr, h_lin, N_NODES, HC, HC);
    gemm(x_cur, Ws1, bs1, skipb, N_NODES, HC, HC);
    attn_logits<<<(nNH + 255) / 256, 256, 0, stream>>>(h_lin, aS1, aD1, als, ald, N_NODES);
    run_edges(h_lin);
    finish01<<<(nNF + 255) / 256, 256, 0, stream>>>(att, b1, skipb, x_cur, N_NODES);

    // ---- layer 2 : HC -> OUT (mean over heads) + L2 normalize ---------------
    gemm(x_cur, W2, nullptr, h_lin, N_NODES, HC, HEADS * OUT_C);
    gemm(x_cur, Ws2, bs2, skipb, N_NODES, HC, OUT_C);
    attn_logits<<<(nNH + 255) / 256, 256, 0, stream>>>(h_lin, aS2, aD2, als, ald, N_NODES);
    run_edges(h_lin);
    finish2<<<(N_NODES + 7) / 8, 256, 0, stream>>>(att, b2, skipb, (float*)d_out, N_NODES);
}